// EndToEndCryptoModel_64338610094577
// MI455X (gfx1250) — compile-verified
//
#include <hip/hip_runtime.h>
#include <hip/hip_bf16.h>
#include <math.h>

// ---------------------------------------------------------------------------
// Model dims (compile-time)
// ---------------------------------------------------------------------------
#define Bm   32
#define Tm   64
#define Nm   128
#define Fm   128
#define Um   64
#define G4   256   // 4*U
#define K1m  64
#define K2m  32
#define EPSf 1e-3f
#define SLP  0.01f

typedef float v2f __attribute__((ext_vector_type(2)));
typedef float v8f __attribute__((ext_vector_type(8)));

// D = A(16x4) * B(4x16) + C   (f32 WMMA, wave32; EXEC must be all ones)
__device__ __forceinline__ v8f wmma4(v2f a, v2f b, v8f c) {
    return __builtin_amdgcn_wmma_f32_16x16x4_f32(false, a, false, b,
                                                 (short)0, c, false, false);
}

__device__ __forceinline__ v8f vzero8() {
    v8f z = {0.f, 0.f, 0.f, 0.f, 0.f, 0.f, 0.f, 0.f};
    return z;
}

__device__ __forceinline__ float sigm(float x) { return 1.f / (1.f + __expf(-x)); }
__device__ __forceinline__ float leaky(float x) { return x >= 0.f ? x : SLP * x; }

// ---------------------------------------------------------------------------
// Kernel 0: zero the 96-float accumulator
// ---------------------------------------------------------------------------
__global__ void k_zero(float* acc) {
    if (threadIdx.x < Bm * 3) acc[threadIdx.x] = 0.f;
}

// ---------------------------------------------------------------------------
// Kernel 1: xz[b*T+t, 256] = x[b*T+t, 128] @ Wk[128,256] + lstm_b
// 2048 rows -> 128 M-tiles x 16 N-tiles = 2048 waves. 8 waves/block.
// ---------------------------------------------------------------------------
__global__ __launch_bounds__(256) void
k_xz(const float* __restrict__ x, const float* __restrict__ Wk,
     const float* __restrict__ bias, float* __restrict__ xz) {
    const int gw   = blockIdx.x * 8 + (threadIdx.x >> 5);
    const int lane = threadIdx.x & 31;
    const int half = lane >> 4, l = lane & 15;
    const int mt = gw >> 4, nt = gw & 15;
    const int m0 = mt * 16, n0 = nt * 16;

    v8f c = vzero8();
    for (int k = 0; k < Fm; k += 4) {
        v2f a, b;
        a[0] = x[(m0 + l) * Fm + k + half * 2 + 0];
        a[1] = x[(m0 + l) * Fm + k + half * 2 + 1];
        b[0] = Wk[(k + half * 2 + 0) * G4 + n0 + l];
        b[1] = Wk[(k + half * 2 + 1) * G4 + n0 + l];
        c = wmma4(a, b, c);
    }
    const float bv = bias[n0 + l];
    for (int r = 0; r < 8; ++r)
        xz[(m0 + r + 8 * half) * G4 + n0 + l] = c[r] + bv;
}

// ---------------------------------------------------------------------------
// Kernel 2: LSTM scan. Each block owns 16 batch rows; h,c,z live in LDS.
// Wr B-fragments are loop-invariant across timesteps: preload all 16 k-step
// fragments (32 VGPRs) once, so the sequential t-loop touches global memory
// only for the xz accumulator init and the hs store.
// ---------------------------------------------------------------------------
__global__ __launch_bounds__(512) void
k_lstm(const float* __restrict__ xz, const float* __restrict__ Wr,
       float* __restrict__ hs) {
    __shared__ float sh[16 * Um];    // h
    __shared__ float sc[16 * Um];    // c
    __shared__ float sz[16 * G4];    // z (pre-activation gates)

    const int bo   = blockIdx.x * 16;
    const int tid  = threadIdx.x;
    const int wave = tid >> 5;         // 0..15 -> N tile
    const int lane = tid & 31;
    const int half = lane >> 4, l = lane & 15;
    const int n0   = wave * 16;

    // Preload the 16 recurrent-weight B fragments for this wave's N tile.
    v2f bw[16];
#pragma unroll
    for (int kk = 0; kk < 16; ++kk) {
        bw[kk][0] = Wr[(kk * 4 + half * 2 + 0) * G4 + n0 + l];
        bw[kk][1] = Wr[(kk * 4 + half * 2 + 1) * G4 + n0 + l];
    }

    for (int i = tid; i < 16 * Um; i += 512) { sh[i] = 0.f; sc[i] = 0.f; }
    __syncthreads();

    for (int t = 0; t < Tm; ++t) {
        // init accumulator from precomputed input projection
        v8f c;
#pragma unroll
        for (int r = 0; r < 8; ++r)
            c[r] = xz[((bo + r + 8 * half) * Tm + t) * G4 + n0 + l];
        // z += h @ Wr   (K = 64)
#pragma unroll
        for (int kk = 0; kk < 16; ++kk) {
            v2f a;
            a[0] = sh[l * Um + kk * 4 + half * 2 + 0];
            a[1] = sh[l * Um + kk * 4 + half * 2 + 1];
            c = wmma4(a, bw[kk], c);
        }
#pragma unroll
        for (int r = 0; r < 8; ++r)
            sz[(r + 8 * half) * G4 + n0 + l] = c[r];
        __syncthreads();

        // gates: 16*64 cells, 2 per thread
        for (int e = tid; e < 16 * Um; e += 512) {
            const int bb = e >> 6, u = e & 63;
            const float ig = sigm(sz[bb * G4 + u]);
            const float fg = sigm(sz[bb * G4 + Um + u]);
            const float gg = tanhf(sz[bb * G4 + 2 * Um + u]);
            const float og = sigm(sz[bb * G4 + 3 * Um + u]);
            const float cc = fg * sc[e] + ig * gg;
            const float hh = og * tanhf(cc);
            sc[e] = cc;
            sh[e] = hh;
            hs[((bo + bb) * Tm + t) * Um + u] = hh;
        }
        __syncthreads();
    }
}

// ---------------------------------------------------------------------------
// Kernel 3: s1 = BN(h) @ w1   [2048,64]@[64,64]; BN fused into A-frag load.
// 128 M-tiles x 4 N-tiles = 512 waves, 8 waves/block.
// ---------------------------------------------------------------------------
__global__ __launch_bounds__(256) void
k_s1(const float* __restrict__ hs, const float* __restrict__ w1,
     const float* __restrict__ bg, const float* __restrict__ bb,
     const float* __restrict__ bmn, const float* __restrict__ bv,
     float* __restrict__ s1) {
    const int gw   = blockIdx.x * 8 + (threadIdx.x >> 5);
    const int lane = threadIdx.x & 31;
    const int half = lane >> 4, l = lane & 15;
    const int mt = gw >> 2, nt = gw & 3;
    const int m0 = mt * 16, n0 = nt * 16;

    v8f c = vzero8();
    for (int k = 0; k < Um; k += 4) {
        const int k0 = k + half * 2, k1 = k0 + 1;
        const float s0  = bg[k0] * rsqrtf(bv[k0] + EPSf);
        const float s1c = bg[k1] * rsqrtf(bv[k1] + EPSf);
        v2f a, b;
        a[0] = (hs[(m0 + l) * Um + k0] - bmn[k0]) * s0 + bb[k0];
        a[1] = (hs[(m0 + l) * Um + k1] - bmn[k1]) * s1c + bb[k1];
        b[0] = w1[k0 * K1m + n0 + l];
        b[1] = w1[k1 * K1m + n0 + l];
        c = wmma4(a, b, c);
    }
    for (int r = 0; r < 8; ++r)
        s1[(m0 + r + 8 * half) * K1m + n0 + l] = c[r];
}

// ---------------------------------------------------------------------------
// Kernel 4: rowsum[b,i] = sum_j a[b,i,j]
// ---------------------------------------------------------------------------
__global__ void k_rowsum(const float* __restrict__ adj, float* __restrict__ rs) {
    const int idx = blockIdx.x * blockDim.x + threadIdx.x;
    if (idx < Bm * Nm) {
        const float* row = adj + (size_t)idx * Nm;
        float s = 0.f;
        for (int j = 0; j < Nm; ++j) s += row[j];
        rs[idx] = s;
    }
}

// ---------------------------------------------------------------------------
// Kernel 5: fused GCN1(broadcast form) + GCN2 + head partial-dot, per (b,t).
// grid = B*T = 2048 blocks of 512 threads (16 waves).
//   g1[i,k]  = BN1(leaky(rowsum[b,i]*s1[b,t,k] + b1[k]))      -> LDS
//   sup2     = g1 @ w2                 [128,64]@[64,32] WMMA  -> LDS
//   g2tile   = a[b] @ sup2             [128,128]@[128,32] WMMA (in regs)
//   g2       = BN2(leaky(g2tile + b2)); acc[b,:] += g2 . d1_w slice
// Adjacency rows + d1_w slice are prefetched (global_prefetch_b8) so they
// stream into cache while the g1/sup2 phases run.
// ---------------------------------------------------------------------------
__global__ __launch_bounds__(512) void
k_gcn(const float* __restrict__ adj, const float* __restrict__ s1,
      const float* __restrict__ rs, const float* __restrict__ b1,
      const float* __restrict__ g1g, const float* __restrict__ g1b,
      const float* __restrict__ g1m, const float* __restrict__ g1v,
      const float* __restrict__ w2, const float* __restrict__ b2,
      const float* __restrict__ g2g, const float* __restrict__ g2b,
      const float* __restrict__ g2m, const float* __restrict__ g2v,
      const float* __restrict__ d1w, float* __restrict__ acc) {
    __shared__ float ls1[K1m];
    __shared__ float lrs[Nm];
    __shared__ float lsc1[K1m], lsh1[K1m];
    __shared__ float lsc2[K2m], lsh2[K2m];
    __shared__ float lg1[Nm * K1m];     // 32 KB
    __shared__ float lsup2[Nm * K2m];   // 16 KB
    __shared__ float lacc[3];

    const int b = blockIdx.x >> 6;
    const int t = blockIdx.x & 63;
    const int tid  = threadIdx.x;
    const int wave = tid >> 5;
    const int lane = tid & 31;
    const int half = lane >> 4, l = lane & 15;
    const int mt = wave >> 1, nt = wave & 1;
    const int m0 = mt * 16, n0 = nt * 16;

    // Prefetch data consumed in the final phase: this wave's adjacency rows
    // and the d1_w slice for (t, this tile). Two prefetch hints per row.
    {
        const float* Ab = adj + (size_t)b * Nm * Nm;
        __builtin_prefetch(&Ab[(m0 + l) * Nm], 0, 1);
        __builtin_prefetch(&Ab[(m0 + l) * Nm + 64], 0, 1);
        const int i = m0 + l;
        __builtin_prefetch(&d1w[((t * Nm + i) * K2m) * 3], 0, 1);
        __builtin_prefetch(&d1w[((t * Nm + i) * K2m + 16) * 3], 0, 1);
    }

    if (tid < K1m) {
        ls1[tid] = s1[((b * Tm + t) * K1m) + tid];
        const float sc = g1g[tid] * rsqrtf(g1v[tid] + EPSf);
        lsc1[tid] = sc;
        lsh1[tid] = g1b[tid] - g1m[tid] * sc;
    }
    if (tid >= 64 && tid < 64 + Nm) lrs[tid - 64] = rs[b * Nm + (tid - 64)];
    if (tid >= 192 && tid < 192 + K2m) {
        const int k = tid - 192;
        const float sc = g2g[k] * rsqrtf(g2v[k] + EPSf);
        lsc2[k] = sc;
        lsh2[k] = g2b[k] - g2m[k] * sc;
    }
    if (tid == 0) { lacc[0] = 0.f; lacc[1] = 0.f; lacc[2] = 0.f; }
    __syncthreads();

    // g1 in LDS (rank-1 outer product + bias, leaky, BN1)
    for (int e = tid; e < Nm * K1m; e += 512) {
        const int i = e >> 6, k = e & 63;
        float vv = lrs[i] * ls1[k] + b1[k];
        vv = leaky(vv);
        lg1[e] = vv * lsc1[k] + lsh1[k];
    }
    __syncthreads();

    // sup2 = g1 @ w2 : 8 M-tiles x 2 N-tiles -> 16 waves
    {
        v8f c = vzero8();
        for (int k = 0; k < K1m; k += 4) {
            v2f a, bf;
            a[0]  = lg1[(m0 + l) * K1m + k + half * 2 + 0];
            a[1]  = lg1[(m0 + l) * K1m + k + half * 2 + 1];
            bf[0] = w2[(k + half * 2 + 0) * K2m + n0 + l];
            bf[1] = w2[(k + half * 2 + 1) * K2m + n0 + l];
            c = wmma4(a, bf, c);
        }
        for (int r = 0; r < 8; ++r)
            lsup2[(m0 + r + 8 * half) * K2m + n0 + l] = c[r];
    }
    __syncthreads();

    // g2 = a[b] @ sup2 ; fuse bias/leaky/BN2 and the d1_w partial dot
    {
        const float* Ab = adj + (size_t)b * Nm * Nm;
        v8f c = vzero8();
        for (int k = 0; k < Nm; k += 4) {
            v2f a, bf;
            a[0]  = Ab[(m0 + l) * Nm + k + half * 2 + 0];
            a[1]  = Ab[(m0 + l) * Nm + k + half * 2 + 1];
            bf[0] = lsup2[(k + half * 2 + 0) * K2m + n0 + l];
            bf[1] = lsup2[(k + half * 2 + 1) * K2m + n0 + l];
            c = wmma4(a, bf, c);
        }
        float p0 = 0.f, p1 = 0.f, p2 = 0.f;
        const int kcol = n0 + l;
        const float bsc = lsc2[kcol], bsh = lsh2[kcol], bias2 = b2[kcol];
        for (int r = 0; r < 8; ++r) {
            const int i = m0 + r + 8 * half;          // node index
            float gv = leaky(c[r] + bias2) * bsc + bsh;
            const int wrow = ((t * Nm + i) * K2m + kcol) * 3;
            p0 += gv * d1w[wrow + 0];
            p1 += gv * d1w[wrow + 1];
            p2 += gv * d1w[wrow + 2];
        }
        atomicAdd(&lacc[0], p0);
        atomicAdd(&lacc[1], p1);
        atomicAdd(&lacc[2], p2);
    }
    __syncthreads();
    if (tid < 3) atomicAdd(&acc[b * 3 + tid], lacc[tid]);
}

// ---------------------------------------------------------------------------
// Kernel 6: out[b,n] = relu(acc[b,:] + d1_b) @ d2_w + d2_b
// ---------------------------------------------------------------------------
__global__ void k_out(const float* __restrict__ acc, const float* __restrict__ d1b,
                      const float* __restrict__ d2w, const float* __restrict__ d2b,
                      float* __restrict__ out) {
    const int idx = blockIdx.x * blockDim.x + threadIdx.x;
    if (idx < Bm * Nm) {
        const int b = idx >> 7, n = idx & 127;
        float o = d2b[n];
        for (int j = 0; j < 3; ++j) {
            float d = acc[b * 3 + j] + d1b[j];
            d = d > 0.f ? d : 0.f;
            o += d * d2w[j * Nm + n];
        }
        out[idx] = o;
    }
}

// ---------------------------------------------------------------------------
extern "C" void kernel_launch(void* const* d_in, const int* in_sizes, int n_in,
                              void* d_out, int out_size, void* d_ws, size_t ws_size,
                              hipStream_t stream) {
    (void)in_sizes; (void)n_in; (void)out_size; (void)ws_size;
    const float* x      = (const float*)d_in[0];
    const float* adj    = (const float*)d_in[1];
    const float* lstm_k = (const float*)d_in[2];
    const float* lstm_r = (const float*)d_in[3];
    const float* lstm_b = (const float*)d_in[4];
    const float* bnl_g  = (const float*)d_in[5];
    const float* bnl_b  = (const float*)d_in[6];
    const float* bnl_m  = (const float*)d_in[7];
    const float* bnl_v  = (const float*)d_in[8];
    const float* w1     = (const float*)d_in[9];
    const float* b1     = (const float*)d_in[10];
    const float* bn1_g  = (const float*)d_in[11];
    const float* bn1_b  = (const float*)d_in[12];
    const float* bn1_m  = (const float*)d_in[13];
    const float* bn1_v  = (const float*)d_in[14];
    const float* w2     = (const float*)d_in[15];
    const float* b2     = (const float*)d_in[16];
    const float* bn2_g  = (const float*)d_in[17];
    const float* bn2_b  = (const float*)d_in[18];
    const float* bn2_m  = (const float*)d_in[19];
    const float* bn2_v  = (const float*)d_in[20];
    const float* d1w    = (const float*)d_in[21];
    const float* d1b    = (const float*)d_in[22];
    const float* d2w    = (const float*)d_in[23];
    const float* d2b    = (const float*)d_in[24];
    float* out = (float*)d_out;

    float* ws = (float*)d_ws;
    float* xz  = ws;                        // 2048*256 = 524288
    float* hs  = ws + 524288;               // 2048*64  = 131072
    float* s1  = ws + 655360;               // 2048*64  = 131072
    float* rs  = ws + 786432;               // 4096
    float* acc = ws + 790528;               // 96

    k_zero<<<1, 128, 0, stream>>>(acc);
    k_xz<<<256, 256, 0, stream>>>(x, lstm_k, lstm_b, xz);
    k_lstm<<<2, 512, 0, stream>>>(xz, lstm_r, hs);
    k_s1<<<64, 256, 0, stream>>>(hs, w1, bnl_g, bnl_b, bnl_m, bnl_v, s1);
    k_rowsum<<<16, 256, 0, stream>>>(adj, rs);
    k_gcn<<<Bm * Tm, 512, 0, stream>>>(adj, s1, rs, b1,
                                       bn1_g, bn1_b, bn1_m, bn1_v,
                                       w2, b2, bn2_g, bn2_b, bn2_m, bn2_v,
                                       d1w, acc);
    k_out<<<16, 256, 0, stream>>>(acc, d1b, d2w, d2b, out);
}